// GNN_50869592654552
// MI455X (gfx1250) — compile-verified
//
#include <hip/hip_runtime.h>
#include <math.h>

typedef float v2f __attribute__((ext_vector_type(2)));
typedef float v8f __attribute__((ext_vector_type(8)));

#define HID 128
#define NUM_TYPES 4

// ---------------- degree / normalization ----------------
__global__ void k_deg_zero(int* __restrict__ deg, int n) {
    int i = blockIdx.x * blockDim.x + threadIdx.x;
    if (i < n) deg[i] = 0;
}

__global__ void k_deg_count(const int* __restrict__ dst, int* __restrict__ deg, int e) {
    int i = blockIdx.x * blockDim.x + threadIdx.x;
    if (i < e) atomicAdd(&deg[dst[i]], 1);
}

__global__ void k_dinv(const int* __restrict__ deg, float* __restrict__ dinv, int n) {
    int i = blockIdx.x * blockDim.x + threadIdx.x;
    if (i < n) dinv[i] = rsqrtf((float)(deg[i] + 1));   // +1 = self loop, deg >= 1
}

// per-edge symmetric norm: norm[e] = dinv[src] * dinv[dst]
__global__ void k_norm(const int* __restrict__ ei, const float* __restrict__ dinv,
                       float* __restrict__ norm, int nE) {
    int e = blockIdx.x * blockDim.x + threadIdx.x;
    if (e < nE) norm[e] = dinv[ei[e]] * dinv[ei[nE + e]];
}

// ---------------- per-type adapter: res = tanh(x @ W[type] + b[type]) ----------------
// One block = 16-row strip (staged in LDS), one wave = 16-col tile.
// FP32 WMMA 16x16x4 keeps full fp32 precision; compute all 4 types (as the
// reference einsum does) and select per row by node_type.
__global__ __launch_bounds__(256)
void k_adapt(const float* __restrict__ x, const int* __restrict__ ntype,
             const float* __restrict__ aW, const float* __restrict__ ab,
             float* __restrict__ res) {
    __shared__ __align__(16) float xs[16 * HID];   // 8 KB strip, reused by all 8 waves
    const int tid = threadIdx.x;
    const int rb  = blockIdx.x * 16;

    {   // cooperative stage: 2048 floats = 512 float4, 2 per thread
        const float4* gsrc = (const float4*)(x + (size_t)rb * HID);
        float4* ldst = (float4*)xs;
        ldst[tid]       = gsrc[tid];
        ldst[tid + 256] = gsrc[tid + 256];
    }
    __syncthreads();

    const int lane = tid & 31;
    const int wave = tid >> 5;
    const int m16  = lane & 15;
    const int g    = lane >> 4;        // half-wave select
    const int col  = wave * 16 + m16;  // output column

    int tr[8];
#pragma unroll
    for (int r = 0; r < 8; ++r) tr[r] = ntype[rb + r + 8 * g];

    v8f sel;
#pragma unroll
    for (int r = 0; r < 8; ++r) sel[r] = 0.0f;

    for (int t = 0; t < NUM_TYPES; ++t) {
        const float* W = aW + (size_t)t * HID * HID;
        v8f acc;
#pragma unroll
        for (int r = 0; r < 8; ++r) acc[r] = 0.0f;

#pragma unroll 4
        for (int k = 0; k < HID; k += 4) {
            // A 16x4 f32: lane holds x[m16, k + 2g + {0,1}]
            v2f a = *(const v2f*)&xs[m16 * HID + k + 2 * g];
            // B 4x16 f32: lane holds W[k + 2g + {0,1}, col]
            v2f b;
            b.x = W[(k + 2 * g) * HID + col];
            b.y = W[(k + 2 * g + 1) * HID + col];
            acc = __builtin_amdgcn_wmma_f32_16x16x4_f32(
                false, a, false, b, (short)0, acc, false, false);
        }
        const float bias = ab[t * HID + col];
#pragma unroll
        for (int r = 0; r < 8; ++r)
            sel[r] = (tr[r] == t) ? (acc[r] + bias) : sel[r];
    }

#pragma unroll
    for (int r = 0; r < 8; ++r)
        res[(size_t)(rb + r + 8 * g) * HID + col] = tanhf(sel[r]);
}

// ---------------- dense GEMM: y = x @ W  (fp32 WMMA) ----------------
__global__ __launch_bounds__(256)
void k_gemm(const float* __restrict__ x, const float* __restrict__ W,
            float* __restrict__ y) {
    __shared__ __align__(16) float xs[16 * HID];
    const int tid = threadIdx.x;
    const int rb  = blockIdx.x * 16;

    {
        const float4* gsrc = (const float4*)(x + (size_t)rb * HID);
        float4* ldst = (float4*)xs;
        ldst[tid]       = gsrc[tid];
        ldst[tid + 256] = gsrc[tid + 256];
    }
    __syncthreads();

    const int lane = tid & 31;
    const int wave = tid >> 5;
    const int m16  = lane & 15;
    const int g    = lane >> 4;
    const int col  = wave * 16 + m16;

    v8f acc;
#pragma unroll
    for (int r = 0; r < 8; ++r) acc[r] = 0.0f;

#pragma unroll 4
    for (int k = 0; k < HID; k += 4) {
        v2f a = *(const v2f*)&xs[m16 * HID + k + 2 * g];
        v2f b;
        b.x = W[(k + 2 * g) * HID + col];
        b.y = W[(k + 2 * g + 1) * HID + col];
        acc = __builtin_amdgcn_wmma_f32_16x16x4_f32(
            false, a, false, b, (short)0, acc, false, false);
    }

#pragma unroll
    for (int r = 0; r < 8; ++r)
        y[(size_t)(rb + r + 8 * g) * HID + col] = acc[r];
}

// ------- out = y * dinv^2 + b  (self-loop + bias + zero-init), float4-wide -------
__global__ void k_init_out(const float4* __restrict__ y4, const float* __restrict__ dinv,
                           const float4* __restrict__ b4, float4* __restrict__ out4,
                           int total4) {
    int i = blockIdx.x * blockDim.x + threadIdx.x;
    if (i < total4) {
        int node = i >> 5;          // 32 float4 per row
        int q    = i & 31;
        float di = dinv[node];
        float s  = di * di;
        float4 v = y4[i];
        float4 bb = b4[q];
        float4 o;
        o.x = v.x * s + bb.x;
        o.y = v.y * s + bb.y;
        o.z = v.z * s + bb.z;
        o.w = v.w * s + bb.w;
        out4[i] = o;
    }
}

// ---------------- edge scatter: out[dst] += y[src] * norm[e] ----------------
// One wave32 per edge. Edge id is forced into an SGPR (readfirstlane) so the
// index/norm loads become scalar loads; each lane moves a float4 of payload.
__global__ __launch_bounds__(256)
void k_scatter(const float* __restrict__ y, const int* __restrict__ ei,
               const float* __restrict__ norm, float* __restrict__ out, int nE) {
    const int waveInBlock = __builtin_amdgcn_readfirstlane(threadIdx.x >> 5);
    const int e = blockIdx.x * 8 + waveInBlock;       // uniform (SGPR)
    if (e >= nE) return;                              // uniform branch
    const int lane = threadIdx.x & 31;

    const int   s   = ei[e];                          // scalar loads (uniform addr)
    const int   d   = ei[nE + e];
    const float nrm = norm[e];

    float4 v = ((const float4*)y)[(size_t)s * 32 + lane];   // global_load_b128
    float* o = out + (size_t)d * HID + lane * 4;
    atomicAdd(o + 0, v.x * nrm);
    atomicAdd(o + 1, v.y * nrm);
    atomicAdd(o + 2, v.z * nrm);
    atomicAdd(o + 3, v.w * nrm);
}

extern "C" void kernel_launch(void* const* d_in, const int* in_sizes, int n_in,
                              void* d_out, int out_size, void* d_ws, size_t ws_size,
                              hipStream_t stream) {
    const float* x     = (const float*)d_in[0];   // node_feature [N,128]
    const int*   ntype = (const int*)d_in[1];     // node_type [N]
    const int*   ei    = (const int*)d_in[3];     // edge_index [2,E]
    const float* aW    = (const float*)d_in[5];   // adapt_W [4,128,128]
    const float* ab    = (const float*)d_in[6];   // adapt_b [4,128]
    const float* W1    = (const float*)d_in[7];
    const float* b1    = (const float*)d_in[8];
    const float* W2    = (const float*)d_in[9];
    const float* b2    = (const float*)d_in[10];
    float* out = (float*)d_out;

    const int N = in_sizes[0] / HID;
    const int E = in_sizes[3] / 2;
    const size_t NF = (size_t)N * HID;

    float* ws   = (float*)d_ws;
    float* res  = ws;                    // adapter output       [N,128]
    float* y    = ws + NF;               // GEMM output (reused) [N,128]
    float* h1   = ws + 2 * NF;           // layer-1 output       [N,128]
    float* dinv = ws + 3 * NF;           // [N]
    int*   deg  = (int*)(ws + 3 * NF + N);       // [N]
    float* norm = ws + 3 * NF + 2 * (size_t)N;   // [E]

    const int T = 256;
    const int total4 = N * (HID / 4);    // float4 elements per field

    // normalization
    k_deg_zero <<<(N + T - 1) / T, T, 0, stream>>>(deg, N);
    k_deg_count<<<(E + T - 1) / T, T, 0, stream>>>(ei + E, deg, E);
    k_dinv     <<<(N + T - 1) / T, T, 0, stream>>>(deg, dinv, N);
    k_norm     <<<(E + T - 1) / T, T, 0, stream>>>(ei, dinv, norm, E);

    // adapter
    k_adapt<<<N / 16, T, 0, stream>>>(x, ntype, aW, ab, res);

    // GCN layer 1
    k_gemm    <<<N / 16, T, 0, stream>>>(res, W1, y);
    k_init_out<<<(total4 + T - 1) / T, T, 0, stream>>>(
        (const float4*)y, dinv, (const float4*)b1, (float4*)h1, total4);
    k_scatter <<<(E + 7) / 8, T, 0, stream>>>(y, ei, norm, h1, E);

    // GCN layer 2
    k_gemm    <<<N / 16, T, 0, stream>>>(h1, W2, y);
    k_init_out<<<(total4 + T - 1) / T, T, 0, stream>>>(
        (const float4*)y, dinv, (const float4*)b2, (float4*)out, total4);
    k_scatter <<<(E + 7) / 8, T, 0, stream>>>(y, ei, norm, out, E);
}